// GAT_LP_12945031430621
// MI455X (gfx1250) — compile-verified
//
#include <hip/hip_runtime.h>
#include <hip/hip_bf16.h>

typedef __attribute__((ext_vector_type(2))) float v2f;
typedef __attribute__((ext_vector_type(8))) float v8f;
typedef __attribute__((ext_vector_type(4))) unsigned int u32x4;
typedef __attribute__((ext_vector_type(8))) int i32x8;
typedef __attribute__((ext_vector_type(4))) int i32x4;

#define HEADS 8

// ---------------------------------------------------------------------------
// Tensor Data Mover: 2D tile (tile_rows x tile_cols f32) from row-major global
// (row stride = row_stride elements) into LDS at byte offset lds_off.
// Descriptor packing per CDNA5 ISA.
// padA: insert 2 dwords of LDS padding after every 32 dwords (row stride 34).
// ---------------------------------------------------------------------------
__device__ __forceinline__ void tdm_load_2d(unsigned lds_off, const float* gptr,
                                            unsigned tile_cols, unsigned tile_rows,
                                            unsigned tensor_cols, unsigned tensor_rows,
                                            unsigned row_stride, bool padA) {
  unsigned long long ga = (unsigned long long)(size_t)gptr;
  u32x4 g0;
  g0[0] = 1u;                                      // count=1, user mode, no gather
  g0[1] = lds_off;                                 // lds_addr (bytes)
  g0[2] = (unsigned)(ga & 0xFFFFFFFFull);          // global_addr[31:0]
  g0[3] = (unsigned)((ga >> 32) & 0x01FFFFFFull)   // global_addr[56:32]
        | (2u << 30);                              // type=2 ("image")
  unsigned d0 = (2u << 16);                        // data_size = 4 bytes
  if (padA) d0 |= (1u << 20)                       // pad_enable
               | (4u << 22)                        // pad_interval: 32 dwords
               | (1u << 25);                       // pad_amount: 2 dwords
  i32x8 g1;
  g1[0] = (int)d0;                                 // wg_mask=0 (not in cluster)
  g1[1] = (int)((tensor_cols & 0xFFFFu) << 16);    // [63:48] tensor_dim0 lo
  g1[2] = (int)((tensor_cols >> 16) | ((tensor_rows & 0xFFFFu) << 16));
  g1[3] = (int)((tensor_rows >> 16) | (tile_cols << 16));   // [127:112] tile_dim0
  g1[4] = (int)tile_rows;                          // tile_dim1 (tile_dim2 = 0)
  g1[5] = (int)row_stride;                         // tensor_dim0_stride[31:0]
  g1[6] = 0;
  g1[7] = 0;
  i32x4 z4 = {0, 0, 0, 0};
  i32x8 z8 = {0, 0, 0, 0, 0, 0, 0, 0};
  __builtin_amdgcn_tensor_load_to_lds(g0, g1, z4, z4, z8, 0);
}

// ---------------------------------------------------------------------------
// WMMA f32 16x16x4 GEMM, TDM-staged through LDS.
// Block = 256 threads = 8 waves as a 4(row) x 2(col) grid of 16x16 tiles
// -> 64x32 output tile per block.  K consumed in 32-wide tiles.
// A fragment (ISA 7.12.2): lane(<16)=row M, VGPRs hold K pair k0+2*half(+1).
// ---------------------------------------------------------------------------
#define APAD 34   // A LDS row stride in dwords (32 data + 2 pad from TDM)

__device__ __forceinline__ void gemm_seg_tdm(v8f& acc, const float* __restrict__ X,
                                             int Kseg, const float* __restrict__ W,
                                             int wrow0, int M, int Nrows,
                                             int row0, int col0,
                                             float* Alds, float* Blds,
                                             unsigned aoff, unsigned boff,
                                             int wave, int lane) {
  const int wr = wave & 3, wc = wave >> 2;
  const int half = lane >> 4, l15 = lane & 15;
  const unsigned trows = (unsigned)(Nrows - row0);   // may be < 64; OOB rows -> 0
  for (int k0 = 0; k0 < Kseg; k0 += 32) {
    __syncthreads();                                 // prev tile fully consumed
    if (wave == 0) {
      tdm_load_2d(aoff, X + (size_t)row0 * Kseg + k0, 32u, 64u,
                  (unsigned)(Kseg - k0), trows, (unsigned)Kseg, true);
      tdm_load_2d(boff, W + (size_t)(wrow0 + k0) * M + col0, 32u, 32u,
                  (unsigned)(M - col0), 32u, (unsigned)M, false);
      __builtin_amdgcn_s_wait_tensorcnt(0);
    }
    __syncthreads();                                 // tiles resident in LDS
#pragma unroll
    for (int ks = 0; ks < 8; ++ks) {
      const int kkl = ks * 4 + 2 * half;
      v2f a = *(const v2f*)&Alds[(wr * 16 + l15) * APAD + kkl];
      v2f b;
      b.x = Blds[kkl * 32 + wc * 16 + l15];
      b.y = Blds[(kkl + 1) * 32 + wc * 16 + l15];
      acc = __builtin_amdgcn_wmma_f32_16x16x4_f32(false, a, false, b, (short)0, acc,
                                                  false, false);
    }
  }
}

__device__ __forceinline__ void store_tile(const v8f& acc, float* __restrict__ Y,
                                           const float* __restrict__ bias,
                                           int Nrows, int M, int row0, int col0,
                                           int wave, int lane) {
  const int half = lane >> 4, l15 = lane & 15;
  const int bcol = col0 + l15;
  const float bb = bias[bcol];
#pragma unroll
  for (int r = 0; r < 8; ++r) {
    const int m = row0 + r + 8 * half;
    if (m < Nrows) Y[(size_t)m * M + bcol] = acc[r] + bb;
  }
}

// Y[Nrows,M] = X[Nrows,K] @ W[K,M] + bias[M]
__global__ __launch_bounds__(256) void wmma_gemm_bias(const float* __restrict__ X,
                                                      const float* __restrict__ W,
                                                      const float* __restrict__ bias,
                                                      float* __restrict__ Y,
                                                      int Nrows, int K, int M) {
  __shared__ float Alds[64 * APAD];
  __shared__ float Blds[32 * 32];
  const unsigned aoff = (unsigned)(size_t)(void*)Alds;
  const unsigned boff = (unsigned)(size_t)(void*)Blds;
  const int lane = threadIdx.x & 31;
  const int wave = threadIdx.x >> 5;
  const int row0 = blockIdx.x * 64 + (wave & 3) * 16;
  const int col0 = blockIdx.y * 32 + (wave >> 2) * 16;
  v8f acc = {};
  gemm_seg_tdm(acc, X, K, W, 0, M, Nrows, blockIdx.x * 64, blockIdx.y * 32,
               Alds, Blds, aoff, boff, wave, lane);
  store_tile(acc, Y, bias, Nrows, M, row0, col0, wave, lane);
}

// JK GEMM: A = concat(X0[N,K0], X1[N,K1], X2[N,K2]) along K; W[(K0+K1+K2),M]
__global__ __launch_bounds__(256) void wmma_gemm_jk(const float* __restrict__ X0, int K0,
                                                    const float* __restrict__ X1, int K1,
                                                    const float* __restrict__ X2, int K2,
                                                    const float* __restrict__ W,
                                                    const float* __restrict__ bias,
                                                    float* __restrict__ Y,
                                                    int Nrows, int M) {
  __shared__ float Alds[64 * APAD];
  __shared__ float Blds[32 * 32];
  const unsigned aoff = (unsigned)(size_t)(void*)Alds;
  const unsigned boff = (unsigned)(size_t)(void*)Blds;
  const int lane = threadIdx.x & 31;
  const int wave = threadIdx.x >> 5;
  const int brow0 = blockIdx.x * 64, bcol0 = blockIdx.y * 32;
  const int row0 = brow0 + (wave & 3) * 16;
  const int col0 = bcol0 + (wave >> 2) * 16;
  v8f acc = {};
  gemm_seg_tdm(acc, X0, K0, W, 0,       M, Nrows, brow0, bcol0, Alds, Blds, aoff, boff, wave, lane);
  gemm_seg_tdm(acc, X1, K1, W, K0,      M, Nrows, brow0, bcol0, Alds, Blds, aoff, boff, wave, lane);
  gemm_seg_tdm(acc, X2, K2, W, K0 + K1, M, Nrows, brow0, bcol0, Alds, Blds, aoff, boff, wave, lane);
  store_tile(acc, Y, bias, Nrows, M, row0, col0, wave, lane);
}

// ---------------------------------------------------------------------------
// Edge phase: GATv2 attention.  One wave per edge; 4 lanes per head.
// ---------------------------------------------------------------------------
__device__ __forceinline__ void edge_endpoints(const int* __restrict__ ei, int E_real,
                                               int e, int& src, int& dst) {
  if (e < E_real) { src = ei[e]; dst = ei[E_real + e]; }
  else            { src = dst = e - E_real; }  // self loops appended
}

__global__ __launch_bounds__(256) void edge_logits_kernel(
    const float* __restrict__ xl, const float* __restrict__ xr,
    const float* __restrict__ att, const int* __restrict__ ei,
    int E_real, int E_tot, int C, int HC,
    float* __restrict__ logits, unsigned int* __restrict__ mEnc) {
  const int e = blockIdx.x * (blockDim.x >> 5) + (threadIdx.x >> 5);
  if (e >= E_tot) return;
  const int lane = threadIdx.x & 31;
  int src, dst;
  edge_endpoints(ei, E_real, e, src, dst);
  const int h = lane >> 2;     // head (8 heads, 4 lanes each)
  const int q = lane & 3;
  const float* pl = xl + (size_t)src * HC + h * C;
  const float* pr = xr + (size_t)dst * HC + h * C;
  const float* pa = att + h * C;
  float s = 0.f;
  for (int c = q; c < C; c += 4) {
    float v = pl[c] + pr[c];
    v = v > 0.f ? v : 0.2f * v;   // leaky_relu(0.2)
    s += v * pa[c];
  }
  s += __shfl_xor(s, 1, 32);
  s += __shfl_xor(s, 2, 32);
  if (q == 0) {
    logits[(size_t)e * HEADS + h] = s;
    unsigned u = __float_as_uint(s);
    u = (u & 0x80000000u) ? ~u : (u | 0x80000000u);  // order-preserving encode
    atomicMax(&mEnc[dst * HEADS + h], u);
  }
}

__global__ __launch_bounds__(256) void edge_exp_kernel(
    const int* __restrict__ ei, int E_real, int E_tot,
    float* __restrict__ logits, const unsigned int* __restrict__ mEnc,
    float* __restrict__ denom) {
  const int t = blockIdx.x * blockDim.x + threadIdx.x;
  if (t >= E_tot * HEADS) return;
  const int e = t >> 3, h = t & 7;
  const int dst = (e < E_real) ? ei[E_real + e] : (e - E_real);
  const unsigned u = mEnc[dst * HEADS + h];
  const float m = (u & 0x80000000u) ? __uint_as_float(u & 0x7FFFFFFFu)
                                    : __uint_as_float(~u);
  const float a = __expf(logits[t] - m);
  logits[t] = a;                     // overwrite logits with exp numerator
  atomicAdd(&denom[dst * HEADS + h], a);
}

__global__ __launch_bounds__(256) void edge_scatter_kernel(
    const float* __restrict__ xl, const float* __restrict__ anum,
    const float* __restrict__ denom, const int* __restrict__ ei,
    int E_real, int E_tot, int HC, int logC, float* __restrict__ agg) {
  const int e = blockIdx.x * (blockDim.x >> 5) + (threadIdx.x >> 5);
  if (e >= E_tot) return;
  const int lane = threadIdx.x & 31;
  int src, dst;
  edge_endpoints(ei, E_real, e, src, dst);
  const float* pn = anum + (size_t)e * HEADS;
  const float* pd = denom + (size_t)dst * HEADS;
  float al[HEADS];
#pragma unroll
  for (int h = 0; h < HEADS; ++h) al[h] = pn[h] / pd[h];
  const float* px = xl + (size_t)src * HC;
  float* po = agg + (size_t)dst * HC;
  for (int i = lane; i < HC; i += 32) {
    atomicAdd(&po[i], al[i >> logC] * px[i]);
  }
}

__global__ __launch_bounds__(256) void elu_bias_kernel(float* __restrict__ agg,
                                                       const float* __restrict__ bias,
                                                       int total, int HCmask) {
  const int t = blockIdx.x * blockDim.x + threadIdx.x;
  if (t >= total) return;
  float v = agg[t] + bias[t & HCmask];
  agg[t] = v > 0.f ? v : (__expf(v) - 1.f);
}

// ---------------------------------------------------------------------------
// Launch
// ---------------------------------------------------------------------------
extern "C" void kernel_launch(void* const* d_in, const int* in_sizes, int n_in,
                              void* d_out, int out_size, void* d_ws, size_t ws_size,
                              hipStream_t stream) {
  const int HID = 128, D1 = 512, D2 = 256, C1 = 64, C2 = 32;
  const int N = in_sizes[0] / HID;
  const int E = in_sizes[1] / 2;
  const int E_tot = E + N;

  const float* x    = (const float*)d_in[0];
  const int*   ei   = (const int*)d_in[1];
  const float* Wl1  = (const float*)d_in[2];
  const float* bl1  = (const float*)d_in[3];
  const float* Wr1  = (const float*)d_in[4];
  const float* br1  = (const float*)d_in[5];
  const float* att1 = (const float*)d_in[6];
  const float* bias1= (const float*)d_in[7];
  const float* Wl2  = (const float*)d_in[8];
  const float* bl2  = (const float*)d_in[9];
  const float* Wr2  = (const float*)d_in[10];
  const float* br2  = (const float*)d_in[11];
  const float* att2 = (const float*)d_in[12];
  const float* bias2= (const float*)d_in[13];
  const float* Wjk  = (const float*)d_in[14];
  const float* bjk  = (const float*)d_in[15];
  float* out = (float*)d_out;

  // Workspace layout (regions A/B reused across layers)
  char* ws = (char*)d_ws;
  const size_t SZ1 = (size_t)N * D1 * sizeof(float);   // 40.96 MB
  const size_t SZ2 = (size_t)N * D2 * sizeof(float);   // 20.48 MB
  const size_t LSZ = (size_t)E_tot * HEADS * sizeof(float);
  const size_t MSZ = (size_t)N * HEADS * sizeof(float);
  float*    xl1   = (float*)(ws);                      // region A
  float*    xr1   = (float*)(ws + SZ1);                // region B
  float*    h1    = (float*)(ws + 2 * SZ1);            // region C (agg1 -> h1)
  float*    logit = (float*)(ws + 3 * SZ1);
  unsigned* mEnc  = (unsigned*)(ws + 3 * SZ1 + LSZ);
  float*    denom = (float*)(ws + 3 * SZ1 + LSZ + MSZ);
  float*    xl2   = xl1;                               // region A reuse
  float*    xr2   = xr1;                               // region B first half
  float*    h2    = (float*)(ws + SZ1 + SZ2);          // region B second half

  const dim3 blk(256);
  const int rowBlocks = (N + 63) / 64;
  const int edgeWaveBlocks = (E_tot + 7) / 8;

  // ---------- Layer 1 ----------
  wmma_gemm_bias<<<dim3(rowBlocks, D1 / 32), blk, 0, stream>>>(x, Wl1, bl1, xl1, N, HID, D1);
  wmma_gemm_bias<<<dim3(rowBlocks, D1 / 32), blk, 0, stream>>>(x, Wr1, br1, xr1, N, HID, D1);

  (void)hipMemsetAsync(mEnc, 0, MSZ, stream);
  (void)hipMemsetAsync(denom, 0, MSZ, stream);
  (void)hipMemsetAsync(h1, 0, SZ1, stream);   // agg1

  edge_logits_kernel<<<edgeWaveBlocks, blk, 0, stream>>>(xl1, xr1, att1, ei, E, E_tot,
                                                         C1, D1, logit, mEnc);
  edge_exp_kernel<<<(E_tot * HEADS + 255) / 256, blk, 0, stream>>>(ei, E, E_tot, logit,
                                                                   mEnc, denom);
  edge_scatter_kernel<<<edgeWaveBlocks, blk, 0, stream>>>(xl1, logit, denom, ei, E, E_tot,
                                                          D1, 6, h1);
  elu_bias_kernel<<<((N * D1) + 255) / 256, blk, 0, stream>>>(h1, bias1, N * D1, D1 - 1);

  // ---------- Layer 2 ----------
  wmma_gemm_bias<<<dim3(rowBlocks, D2 / 32), blk, 0, stream>>>(h1, Wl2, bl2, xl2, N, D1, D2);
  wmma_gemm_bias<<<dim3(rowBlocks, D2 / 32), blk, 0, stream>>>(h1, Wr2, br2, xr2, N, D1, D2);

  (void)hipMemsetAsync(mEnc, 0, MSZ, stream);
  (void)hipMemsetAsync(denom, 0, MSZ, stream);
  (void)hipMemsetAsync(h2, 0, SZ2, stream);   // agg2

  edge_logits_kernel<<<edgeWaveBlocks, blk, 0, stream>>>(xl2, xr2, att2, ei, E, E_tot,
                                                         C2, D2, logit, mEnc);
  edge_exp_kernel<<<(E_tot * HEADS + 255) / 256, blk, 0, stream>>>(ei, E, E_tot, logit,
                                                                   mEnc, denom);
  edge_scatter_kernel<<<edgeWaveBlocks, blk, 0, stream>>>(xl2, logit, denom, ei, E, E_tot,
                                                          D2, 5, h2);
  elu_bias_kernel<<<((N * D2) + 255) / 256, blk, 0, stream>>>(h2, bias2, N * D2, D2 - 1);

  // ---------- JumpingKnowledge cat + final linear ----------
  wmma_gemm_jk<<<dim3(rowBlocks, HID / 32), blk, 0, stream>>>(x, HID, h1, D1, h2, D2,
                                                              Wjk, bjk, out, N, HID);
}